// GCN_22033182228604
// MI455X (gfx1250) — compile-verified
//
#include <hip/hip_runtime.h>
#include <cstdint>
#include <cstddef>

// ---------------------------------------------------------------------------
// GCN (3-layer, GCNConv + BN + ReLU + log_softmax) for gfx1250 (MI455X).
// Dense GEMMs use V_WMMA_F32_16X16X4_F32 (fp32 matrix pipe, matches reference
// precision). Edge aggregation is the memory-bound part -> fp32 HW atomics.
// b0/b1 are skipped: BatchNorm's mean subtraction cancels them exactly.
// ---------------------------------------------------------------------------

typedef float v2f __attribute__((ext_vector_type(2)));
typedef float v8f __attribute__((ext_vector_type(8)));

#define BN_EPS 1e-5f

__device__ __forceinline__ void atomAddF(float* p, float v) {
  __hip_atomic_fetch_add(p, v, __ATOMIC_RELAXED, __HIP_MEMORY_SCOPE_AGENT);
}

// ---------------- small utility kernels ----------------

__global__ void k_fill(float* __restrict__ p, long long n, float val) {
  long long i = blockIdx.x * (long long)blockDim.x + threadIdx.x;
  if (i < n) p[i] = val;
}

__global__ void k_count_deg(const long long* __restrict__ dst,
                            float* __restrict__ deg, long long E) {
  long long i = blockIdx.x * (long long)blockDim.x + threadIdx.x;
  if (i < E) atomAddF(&deg[dst[i]], 1.0f);
}

__global__ void k_rsqrt_ip(float* __restrict__ p, long long n) {
  long long i = blockIdx.x * (long long)blockDim.x + threadIdx.x;
  if (i < n) p[i] = rsqrtf(p[i]);   // deg >= 1 always (self loops)
}

// ---------------- WMMA fp32 GEMM: C[M x Np] = A[M x K] @ B[K x ldb] ----------------
// Np = padded output cols (multiple of 16); ldb = true cols of B (<= Np).
// One wave computes one 16x16 tile via V_WMMA_F32_16X16X4_F32.

__global__ void __launch_bounds__(256)
k_gemm_wmma(const float* __restrict__ A, const float* __restrict__ B,
            float* __restrict__ C, int M, int K, int Np, int ldb) {
  int wave = (int)((blockIdx.x * (long long)blockDim.x + threadIdx.x) >> 5);
  int lane = threadIdx.x & 31;
  int tilesN = Np >> 4;
  int tiles  = (M >> 4) * tilesN;
  if (wave >= tiles) return;                   // wave-uniform: EXEC stays all-1s
  int tm = wave / tilesN;
  int tn = wave - tm * tilesN;

  int  laneLo = lane & 15;
  bool hi     = lane >= 16;
  int  koff   = hi ? 2 : 0;

  // A: 16x4 fp32 tile. lane<16: {K=k,k+1}; lane>=16: {K=k+2,k+3} (ISA layout).
  const float* Arow = A + (size_t)(tm * 16 + laneLo) * K + koff;

  // B: 4x16 fp32 tile, rows striped across lanes; mask cols >= ldb (layer 3 pad)
  // via address clamp + zero-multiply (no divergent EXEC around WMMA).
  int   bc    = tn * 16 + laneLo;
  float bmask = (bc < ldb) ? 1.0f : 0.0f;
  int   bcc   = (bc < ldb) ? bc : (ldb - 1);

  v8f acc = {0.f, 0.f, 0.f, 0.f, 0.f, 0.f, 0.f, 0.f};
  for (int k = 0; k < K; k += 4) {
    v2f a;
    a.x = Arow[k];
    a.y = Arow[k + 1];
    v2f b;
    b.x = B[(size_t)(k + koff)     * ldb + bcc] * bmask;
    b.y = B[(size_t)(k + koff + 1) * ldb + bcc] * bmask;
    acc = __builtin_amdgcn_wmma_f32_16x16x4_f32(
        /*neg_a=*/false, a, /*neg_b=*/false, b,
        /*c_mod=*/(short)0, acc, /*reuse_a=*/false, /*reuse_b=*/false);
  }

  // C/D layout: VGPR r -> row r (lanes 0-15) / row r+8 (lanes 16-31).
  float* Cp = C + (size_t)(tm * 16 + (hi ? 8 : 0)) * Np + tn * 16 + laneLo;
#pragma unroll
  for (int r = 0; r < 8; ++r) Cp[(size_t)r * Np] = acc[r];
}

// ---------------- aggregation (self loop init + edge scatter) ----------------

// out[i, :] = h[i, :] * dinv[i]^2   (self-loop term; also initializes out)
__global__ void k_agg_self(const float4* __restrict__ h4,
                           const float* __restrict__ dinv,
                           float4* __restrict__ out4,
                           long long total, int Hq) {
  long long tid = blockIdx.x * (long long)blockDim.x + threadIdx.x;
  if (tid >= total) return;
  long long row = tid / Hq;
  float c = dinv[row];
  c *= c;
  float4 v = h4[tid];
  out4[tid] = make_float4(v.x * c, v.y * c, v.z * c, v.w * c);
}

// out[dst, :] += h[src, :] * dinv[src]*dinv[dst]; Hq = H/4 threads per edge.
__global__ void k_agg_edges(const float4* __restrict__ h4,
                            const long long* __restrict__ src,
                            const long long* __restrict__ dst,
                            const float* __restrict__ dinv,
                            float* __restrict__ out,
                            long long E, int Hq, int H) {
  long long tid = blockIdx.x * (long long)blockDim.x + threadIdx.x;
  long long e = tid / Hq;
  if (e >= E) return;
  int f4 = (int)(tid - e * Hq);
  long long s = src[e], d = dst[e];
  float coef = dinv[s] * dinv[d];
  float4 v = h4[s * (long long)Hq + f4];
  float* o = out + d * (long long)H + (long long)f4 * 4;
  atomAddF(o + 0, v.x * coef);
  atomAddF(o + 1, v.y * coef);
  atomAddF(o + 2, v.z * coef);
  atomAddF(o + 3, v.w * coef);
}

// ---------------- batch-norm statistics + apply ----------------

// blockDim.x == H (256). Each block accumulates a chunk of rows per column.
__global__ void k_col_stats(const float* __restrict__ h,
                            float* __restrict__ sums,   // [2*H]: sum, sumsq
                            long long Nn, int H, int rowsPerBlock) {
  int col = threadIdx.x;
  long long r0 = (long long)blockIdx.x * rowsPerBlock;
  long long rend = Nn - r0;
  if (rend > rowsPerBlock) rend = rowsPerBlock;
  const float* p = h + r0 * H + col;
  float s = 0.f, s2 = 0.f;
  for (long long r = 0; r < rend; ++r) {
    float v = p[r * H];
    s += v;
    s2 += v * v;
  }
  atomAddF(&sums[col], s);
  atomAddF(&sums[H + col], s2);
}

// y = relu((x - mean) * rsqrt(var + eps) * gamma + beta), in place, float4.
__global__ void k_bn_relu(float4* __restrict__ h4,
                          const float* __restrict__ sums,
                          const float* __restrict__ gamma,
                          const float* __restrict__ beta,
                          long long total, int Hq, int H, float invN) {
  long long tid = blockIdx.x * (long long)blockDim.x + threadIdx.x;
  if (tid >= total) return;
  int c4 = (int)(tid % Hq) * 4;
  float4 v = h4[tid];
  float o[4];
  float vv[4] = {v.x, v.y, v.z, v.w};
#pragma unroll
  for (int j = 0; j < 4; ++j) {
    int c = c4 + j;
    float mean = sums[c] * invN;
    float var  = sums[H + c] * invN - mean * mean;
    float y = (vv[j] - mean) * rsqrtf(var + BN_EPS) * gamma[c] + beta[c];
    o[j] = fmaxf(y, 0.f);
  }
  h4[tid] = make_float4(o[0], o[1], o[2], o[3]);
}

// ---------------- fused bias + log_softmax (C=47, padded stride 64) ----------------

__global__ void k_logsoftmax47(const float* __restrict__ h,  // [N x 64], cols 47..63 zero
                               const float* __restrict__ b2,
                               float* __restrict__ out,      // [N x 47]
                               long long Nn) {
  long long wave = (blockIdx.x * (long long)blockDim.x + threadIdx.x) >> 5;
  int lane = threadIdx.x & 31;
  if (wave >= Nn) return;
  const float* p = h + wave * 64;
  float v0 = (lane < 47)      ? p[lane]      + b2[lane]      : -3.4e38f;
  float v1 = (lane + 32 < 47) ? p[lane + 32] + b2[lane + 32] : -3.4e38f;
  float m = fmaxf(v0, v1);
#pragma unroll
  for (int o = 16; o > 0; o >>= 1) m = fmaxf(m, __shfl_xor(m, o, 32));
  float s = 0.f;
  if (lane < 47)      s += __expf(v0 - m);
  if (lane + 32 < 47) s += __expf(v1 - m);
#pragma unroll
  for (int o = 16; o > 0; o >>= 1) s += __shfl_xor(s, o, 32);
  float lse = m + __logf(s);
  if (lane < 47)      out[wave * 47 + lane]      = v0 - lse;
  if (lane + 32 < 47) out[wave * 47 + lane + 32] = v1 - lse;
}

// ---------------- host launcher ----------------

static inline size_t alignUp(size_t x, size_t a) { return (x + a - 1) & ~(a - 1); }

extern "C" void kernel_launch(void* const* d_in, const int* in_sizes, int n_in,
                              void* d_out, int out_size, void* d_ws, size_t ws_size,
                              hipStream_t stream) {
  const int F = 128, H = 256, Cc = 47, CP = 64;  // CP = padded class dim
  const float*     x   = (const float*)d_in[0];
  const long long* ei  = (const long long*)d_in[1];   // [2, E] int64
  const float*     W0  = (const float*)d_in[3];
  // d_in[4] = b0 (cancels under BN), d_in[2] = relations (unused by reference)
  const float*     g0  = (const float*)d_in[5];
  const float*     be0 = (const float*)d_in[6];
  const float*     W1  = (const float*)d_in[7];
  // d_in[8] = b1 (cancels under BN)
  const float*     g1  = (const float*)d_in[9];
  const float*     be1 = (const float*)d_in[10];
  const float*     W2  = (const float*)d_in[11];
  const float*     b2  = (const float*)d_in[12];

  const long long Nn = (long long)in_sizes[0] / F;
  const long long E  = (long long)in_sizes[1] / 2;
  const long long* srcp = ei;
  const long long* dstp = ei + E;

  // workspace layout
  char* w = (char*)d_ws;
  size_t off = 0;
  float* dinv  = (float*)(w + off); off = alignUp(off + (size_t)Nn * 4, 256);
  float* stats = (float*)(w + off); off = alignUp(off + 2 * H * 4, 256);
  float* buf1  = (float*)(w + off); off = alignUp(off + (size_t)Nn * H * 4, 256);
  float* buf2  = (float*)(w + off); off = alignUp(off + (size_t)Nn * H * 4, 256);
  float* h3    = buf1;                          // [N x 64] (reuses buf1)
  float* agg3  = buf1 + (size_t)Nn * CP;        // [N x 64] (also inside buf1)
  (void)ws_size; (void)n_in; (void)out_size;

  const int B = 256;
  auto blocks = [](long long total, int b) { return (unsigned)((total + b - 1) / b); };

  // ---- degree / dinv ----
  hipLaunchKernelGGL(k_fill, dim3(blocks(Nn, B)), dim3(B), 0, stream, dinv, Nn, 1.0f);
  hipLaunchKernelGGL(k_count_deg, dim3(blocks(E, B)), dim3(B), 0, stream, dstp, dinv, E);
  hipLaunchKernelGGL(k_rsqrt_ip, dim3(blocks(Nn, B)), dim3(B), 0, stream, dinv, Nn);

  const long long tilesL12 = (Nn >> 4) * (H >> 4);   // waves for H=256 GEMMs
  const long long tilesL3  = (Nn >> 4) * (CP >> 4);
  const long long elemsH   = Nn * (H / 4);           // float4 elements, H=256
  const long long elemsC   = Nn * (CP / 4);
  const float invN = 1.0f / (float)Nn;

  // ---- layer 0: GEMM (N x 128)@(128 x 256) -> buf1, aggregate -> buf2 ----
  hipLaunchKernelGGL(k_gemm_wmma, dim3(blocks(tilesL12, 8)), dim3(B), 0, stream,
                     x, W0, buf1, (int)Nn, F, H, H);
  hipLaunchKernelGGL(k_agg_self, dim3(blocks(elemsH, B)), dim3(B), 0, stream,
                     (const float4*)buf1, dinv, (float4*)buf2, elemsH, H / 4);
  hipLaunchKernelGGL(k_agg_edges, dim3(blocks(E * (H / 4), B)), dim3(B), 0, stream,
                     (const float4*)buf1, srcp, dstp, dinv, buf2, E, H / 4, H);
  // ---- BN + ReLU (b0 cancels) ----
  hipLaunchKernelGGL(k_fill, dim3(blocks(2 * H, B)), dim3(B), 0, stream, stats, (long long)(2 * H), 0.0f);
  hipLaunchKernelGGL(k_col_stats, dim3(blocks(Nn, 256)), dim3(H), 0, stream,
                     buf2, stats, Nn, H, 256);
  hipLaunchKernelGGL(k_bn_relu, dim3(blocks(elemsH, B)), dim3(B), 0, stream,
                     (float4*)buf2, stats, g0, be0, elemsH, H / 4, H, invN);

  // ---- layer 1: GEMM (N x 256)@(256 x 256) -> buf1, aggregate -> buf2 ----
  hipLaunchKernelGGL(k_gemm_wmma, dim3(blocks(tilesL12, 8)), dim3(B), 0, stream,
                     buf2, W1, buf1, (int)Nn, H, H, H);
  hipLaunchKernelGGL(k_agg_self, dim3(blocks(elemsH, B)), dim3(B), 0, stream,
                     (const float4*)buf1, dinv, (float4*)buf2, elemsH, H / 4);
  hipLaunchKernelGGL(k_agg_edges, dim3(blocks(E * (H / 4), B)), dim3(B), 0, stream,
                     (const float4*)buf1, srcp, dstp, dinv, buf2, E, H / 4, H);
  hipLaunchKernelGGL(k_fill, dim3(blocks(2 * H, B)), dim3(B), 0, stream, stats, (long long)(2 * H), 0.0f);
  hipLaunchKernelGGL(k_col_stats, dim3(blocks(Nn, 256)), dim3(H), 0, stream,
                     buf2, stats, Nn, H, 256);
  hipLaunchKernelGGL(k_bn_relu, dim3(blocks(elemsH, B)), dim3(B), 0, stream,
                     (float4*)buf2, stats, g1, be1, elemsH, H / 4, H, invN);

  // ---- layer 2: GEMM (N x 256)@(256 x 47->64) -> h3, aggregate -> agg3 ----
  hipLaunchKernelGGL(k_gemm_wmma, dim3(blocks(tilesL3, 8)), dim3(B), 0, stream,
                     buf2, W2, h3, (int)Nn, H, CP, Cc);
  hipLaunchKernelGGL(k_agg_self, dim3(blocks(elemsC, B)), dim3(B), 0, stream,
                     (const float4*)h3, dinv, (float4*)agg3, elemsC, CP / 4);
  hipLaunchKernelGGL(k_agg_edges, dim3(blocks(E * (CP / 4), B)), dim3(B), 0, stream,
                     (const float4*)h3, srcp, dstp, dinv, agg3, E, CP / 4, CP);

  // ---- bias + log_softmax -> d_out [N x 47] ----
  hipLaunchKernelGGL(k_logsoftmax47, dim3(blocks(Nn * 32, B)), dim3(B), 0, stream,
                     agg3, b2, (float*)d_out, Nn);
}